// PolyformerEmbedding_51264729645058
// MI455X (gfx1250) — compile-verified
//
#include <hip/hip_runtime.h>

typedef __attribute__((ext_vector_type(2))) float v2f;
typedef __attribute__((ext_vector_type(4))) float v4f;
typedef __attribute__((ext_vector_type(8))) float v8f;

#define MROW 32        // matrix size (rows i, samples j)
#define NCOEF 8        // DEGREE+1
#define NPAD 16        // coeffs padded to WMMA N dim
#define SEQ 1024
#define BTILE 16       // batch rows per block
#define PAD_STRIDE 1028  // 1024 + 4 dwords pad: bank-conflict-free, 16B aligned rows

// ---------------------------------------------------------------------------
// Setup: per row i (32 threads), build Chebyshev Vandermonde, Gram matrix,
// invert 8x8 via Gauss-Jordan, emit Wt[i][j][d] = (G^-1 V^T)[d][j], padded
// with zeros to 16 "coeff" columns so the main kernel loads unconditionally.
// ---------------------------------------------------------------------------
__global__ __launch_bounds__(32) void cheb_setup(const float* __restrict__ tg,
                                                 float* __restrict__ wt) {
    const int i = threadIdx.x;  // row 0..31

    float G[NCOEF][NCOEF];
    for (int d = 0; d < NCOEF; ++d)
        for (int e = 0; e < NCOEF; ++e) G[d][e] = 0.0f;

    for (int j = 0; j < MROW; ++j) {
        const float t = tg[j * MROW + i];           // times[i][j] = tg[j*32 + i]
        float c[NCOEF];
        c[0] = 1.0f; c[1] = t;
        for (int d = 2; d < NCOEF; ++d) c[d] = 2.0f * t * c[d - 1] - c[d - 2];
        for (int d = 0; d < NCOEF; ++d)
            for (int e = 0; e < NCOEF; ++e) G[d][e] += c[d] * c[e];
    }

    // Gauss-Jordan: [G | I] -> [I | G^-1]  (Chebyshev Gram is well-conditioned)
    float A[NCOEF][2 * NCOEF];
    for (int d = 0; d < NCOEF; ++d) {
        for (int e = 0; e < NCOEF; ++e) {
            A[d][e] = G[d][e];
            A[d][NCOEF + e] = (d == e) ? 1.0f : 0.0f;
        }
    }
    for (int p = 0; p < NCOEF; ++p) {
        const float inv = 1.0f / A[p][p];
        for (int c2 = 0; c2 < 2 * NCOEF; ++c2) A[p][c2] *= inv;
        for (int r = 0; r < NCOEF; ++r) {
            if (r == p) continue;
            const float f = A[r][p];
            for (int c2 = 0; c2 < 2 * NCOEF; ++c2) A[r][c2] -= f * A[p][c2];
        }
    }

    for (int j = 0; j < MROW; ++j) {
        const float t = tg[j * MROW + i];
        float c[NCOEF];
        c[0] = 1.0f; c[1] = t;
        for (int d = 2; d < NCOEF; ++d) c[d] = 2.0f * t * c[d - 1] - c[d - 2];
        float* o = &wt[(size_t)(i * MROW + j) * NPAD];
        for (int d = 0; d < NCOEF; ++d) {
            float w = 0.0f;
            for (int e = 0; e < NCOEF; ++e) w += A[d][NCOEF + e] * c[e];
            o[d] = w;
        }
        for (int d = NCOEF; d < NPAD; ++d) o[d] = 0.0f;
    }
}

// ---------------------------------------------------------------------------
// Main: block = 256 threads (8 wave32), 16 batch rows per block.
// Stage x tile into LDS (coalesced, non-temporal 128b loads: 256 MB read-once
// stream must not thrash the 192 MB L2 / evict the hot W table), then per row
// i compute C[16 batch x 16 coef] = X[16 x 32] * Wt[i] (32 x 16) as 8 chained
// v_wmma_f32_16x16x4_f32 (full f32 precision; kernel is HBM-bound).
// ---------------------------------------------------------------------------
__global__ __launch_bounds__(256) void cheb_fit(const float* __restrict__ x,
                                                const float* __restrict__ wt,
                                                float* __restrict__ out) {
    __shared__ float xt[BTILE * PAD_STRIDE];

    const int tid = threadIdx.x;
    const int b0  = blockIdx.x * BTILE;

    // Stage 16 batch rows (16 * 4 KB) into LDS: global_load_b128 th:NT.
    #pragma unroll 4
    for (int r = 0; r < BTILE; ++r) {
        const v4f* src = (const v4f*)(x + (size_t)(b0 + r) * SEQ);
        const v4f v = __builtin_nontemporal_load(src + tid);
        *(v4f*)(&xt[r * PAD_STRIDE + tid * 4]) = v;
    }
    __syncthreads();

    const int wave = tid >> 5;
    const int lane = tid & 31;
    const int half = lane >> 4;  // 0: lanes 0-15, 1: lanes 16-31
    const int m    = lane & 15;  // A: batch row; B: coeff col; C: N index

    #pragma unroll
    for (int rr = 0; rr < 4; ++rr) {
        const int i = wave * 4 + rr;           // tile row handled by this wave
        v8f c = {0.f, 0.f, 0.f, 0.f, 0.f, 0.f, 0.f, 0.f};

        const float* xrow = &xt[m * PAD_STRIDE + i];        // x_tile[m][i + 32*k]
        const float* wrow = &wt[(size_t)i * MROW * NPAD];   // Wt[i][k][n]

        #pragma unroll
        for (int kc = 0; kc < 8; ++kc) {
            // 32-bit A 16x4 layout: vgpr0 = K0 (lanes 0-15) / K2 (lanes 16-31),
            //                       vgpr1 = K1 / K3  -> lane holds K = 4kc+2*half+{0,1}
            const int k0 = kc * 4 + half * 2;
            v2f a, b;
            a.x = xrow[(k0 + 0) * MROW];
            a.y = xrow[(k0 + 1) * MROW];
            // B 4x16 row-striped-across-lanes, mirrored K interleave
            b.x = wrow[(k0 + 0) * NPAD + m];
            b.y = wrow[(k0 + 1) * NPAD + m];
            c = __builtin_amdgcn_wmma_f32_16x16x4_f32(
                    false, a, false, b, (short)0, c, false, false);
        }

        // C/D layout: vgpr r -> (M = r + 8*half, N = m). Valid coeffs: N < 8.
        // Write-once output stream -> non-temporal stores.
        if (m < NCOEF) {
            #pragma unroll
            for (int r = 0; r < 8; ++r) {
                const int bb = b0 + r + half * 8;
                __builtin_nontemporal_store(
                    c[r], &out[(size_t)bb * (MROW * NCOEF) + i * NCOEF + m]);
            }
        }
    }
}

extern "C" void kernel_launch(void* const* d_in, const int* in_sizes, int n_in,
                              void* d_out, int out_size, void* d_ws, size_t ws_size,
                              hipStream_t stream) {
    const float* x    = (const float*)d_in[0];  // [65536, 1024]
    const float* tg   = (const float*)d_in[1];  // [1024]
    float*       out  = (float*)d_out;          // [65536, 32, 8]
    float*       wtab = (float*)d_ws;           // 32*32*16 floats = 64 KB

    cheb_setup<<<1, 32, 0, stream>>>(tg, wtab);

    const int nblocks = 65536 / BTILE;          // 4096
    cheb_fit<<<nblocks, 256, 0, stream>>>(x, wtab, out);
}